// SegmenterTensorFlow_28698971472345
// MI455X (gfx1250) — compile-verified
//
#include <hip/hip_runtime.h>
#include <stdint.h>

// SegmenterTensorFlow round-trip: segment (windowed framing) + overlap-add.
// Collapses to y[b,n] = x[b,n] * W(n); W periodic with period HOP except the
// first/last SEG-HOP samples of each row. Streaming ~268 MB @ 23.3 TB/s
// => ~11.5 us floor. Grid is sized to exactly one row (N/1024 blocks x 256
// threads x float4), so each thread's weight vector is computed ONCE and the
// inner loop over the batch dimension is pure load-mul-store.
// CDNA5 paths: async global->LDS window staging (global_load_async_to_lds_b128
// + s_wait_asynccnt), NT-hinted b128 streaming, LDS b128 weight load, wave32.

#define HOP 256
#define SEG 1024

typedef float v4f __attribute__((ext_vector_type(4)));

__global__ __launch_bounds__(256) void seg_ola_kernel(
    const float* __restrict__ x,
    const float* __restrict__ aw,
    const float* __restrict__ sw,
    float* __restrict__ y,
    unsigned quarterN,    // N/4 : float4 chunks per row
    unsigned nrows,       // B
    unsigned interior_hi, // nseg * HOP : first n of the tail edge region
    unsigned nseg)        // number of segments
{
    __shared__ __align__(16) float s_aw[SEG];
    __shared__ __align__(16) float s_sw[SEG];
    __shared__ __align__(16) float s_c[SEG];    // aw[j]*sw[j]
    __shared__ __align__(16) float s_wper[HOP]; // interior periodic weight

    const unsigned t = threadIdx.x; // 0..255

    // ---- Stage both 4 KB windows into LDS via CDNA5 async global->LDS DMA.
    {
        uint32_t ldsa = (uint32_t)(uintptr_t)(&s_aw[0]) + t * 16u;
        uint32_t ldss = (uint32_t)(uintptr_t)(&s_sw[0]) + t * 16u;
        const char* ga = (const char*)aw + t * 16u;
        const char* gs = (const char*)sw + t * 16u;
        asm volatile("global_load_async_to_lds_b128 %0, %1, off"
                     :: "v"(ldsa), "v"(ga) : "memory");
        asm volatile("global_load_async_to_lds_b128 %0, %1, off"
                     :: "v"(ldss), "v"(gs) : "memory");
#if __has_builtin(__builtin_amdgcn_s_wait_asynccnt)
        __builtin_amdgcn_s_wait_asynccnt(0);
#else
        asm volatile("s_wait_asynccnt 0" ::: "memory");
#endif
    }
    __syncthreads();

    // c[j] = aw[j] * sw[j]
#pragma unroll
    for (int m2 = 0; m2 < SEG / 256; ++m2) {
        unsigned j = t + (unsigned)m2 * 256u;
        s_c[j] = s_aw[j] * s_sw[j];
    }
    __syncthreads();

    // Interior periodic weight: all 4 overlapping frames present.
    s_wper[t] = s_c[t] + s_c[t + 256] + s_c[t + 512] + s_c[t + 768];
    __syncthreads();

    // ---- Per-thread in-row position: LOOP-INVARIANT across the batch dim.
    const unsigned base4 = blockIdx.x * 256u + t; // float4 index within a row
    if (base4 >= quarterN) return;
    const unsigned n = base4 << 2;                // sample index within a row

    v4f wv;
    if (n >= (unsigned)(SEG - HOP) && n < interior_hi) {
        // Interior: n%4==0 => r%4==0, aligned LDS b128 load, done once.
        wv = *(const v4f*)(&s_wper[n & (HOP - 1u)]);
    } else {
        // Edge (first/last 768 samples of a row): bounded 4-term sum, once.
#pragma unroll
        for (int e = 0; e < 4; ++e) {
            const unsigned ne = n + (unsigned)e;
            const unsigned r  = ne & (HOP - 1u);
            const unsigned q  = ne >> 8; // ne / HOP
            float w = 0.0f;
#pragma unroll
            for (unsigned m = 0; m < 4; ++m) {
                if (q >= m && (q - m) < nseg)
                    w += s_c[r + m * HOP];
            }
            wv[e] = w;
        }
    }

    // ---- Pure streaming row loop: NT b128 load -> mul -> NT b128 store.
    const v4f* xp = (const v4f*)x + base4;
    v4f*       yp = (v4f*)y + base4;
#pragma unroll 4
    for (unsigned b = 0; b < nrows; ++b) {
        v4f xv = __builtin_nontemporal_load(xp);
        __builtin_nontemporal_store(xv * wv, yp);
        xp += quarterN;
        yp += quarterN;
    }
}

extern "C" void kernel_launch(void* const* d_in, const int* in_sizes, int n_in,
                              void* d_out, int out_size, void* d_ws, size_t ws_size,
                              hipStream_t stream) {
    (void)n_in; (void)d_out; (void)d_ws; (void)ws_size;
    const float* x  = (const float*)d_in[0];
    const float* aw = (const float*)d_in[1];
    const float* sw = (const float*)d_in[2];
    float* y = (float*)d_out;

    const unsigned B = 16u;
    const unsigned N = (unsigned)in_sizes[0] / B;           // 2^21
    const unsigned nseg = (N - SEG) / HOP + 1u;             // 8189
    const unsigned interior_hi = nseg * HOP;                // 2096384
    const unsigned quarterN = N / 4u;                       // 524288
    (void)out_size;

    const unsigned blocks = (quarterN + 255u) / 256u;       // 2048: exactly 1 row
    seg_ola_kernel<<<dim3(blocks), dim3(256), 0, stream>>>(
        x, aw, sw, y, quarterN, B, interior_hi, nseg);
}